// FlowMatching_51831665328533
// MI455X (gfx1250) — compile-verified
//
#include <hip/hip_runtime.h>
#include <math.h>

// ---------------------------------------------------------------------------
// FlowMatching blur+lerp, fused single pass for gfx1250 (MI455X).
//   out = t*GaussBlur_sigma(t)(x) + (1-t)*x ,  sigma = 0.1 + 0.9*t[b]
// Separable 9-tap conv executed as banded GEMMs on V_WMMA_F32_16X16X4_F32.
// Memory-bound: ~806MB HBM traffic -> ~35us floor at 23.3 TB/s; b128
// staging/writeout keeps instruction issue below the per-WGP HBM time.
// ---------------------------------------------------------------------------

#define TILE     64
#define HALO     4
#define TAPS     9
#define IN_ROWS  80          // TILE + 16: whole number of 16-row WMMA chunks
#define IN_COLS  72          // TILE + 2*HALO = 18 float4 per row, 16B aligned
#define IN_PITCH 76          // 76*4 and 4*4 are 16B-multiples -> b128 legal
#define H_PITCH  72          // 2*72 mod 64 == 16 -> lane halves hit disjoint banks

typedef __attribute__((ext_vector_type(2))) float v2f;
typedef __attribute__((ext_vector_type(4))) float v4f;
typedef __attribute__((ext_vector_type(8))) float v8f;

__device__ __forceinline__ v8f wmma4(v2f a, v2f b, v8f c) {
  // D(16x16,f32) += A(16x4,f32) * B(4x16,f32)
  return __builtin_amdgcn_wmma_f32_16x16x4_f32(
      /*neg_a=*/false, a, /*neg_b=*/false, b,
      /*c_mod=*/(short)0, c, /*reuse_a=*/false, /*reuse_b=*/false);
}

// Per-batch normalized 1-D taps. The reference's 2D kernel is an exact outer
// product of these (mask is separable, normalizer = (sum1d)^2).
__global__ void gauss_taps_kernel(const float* __restrict__ t,
                                  float* __restrict__ w, int B) {
  int b = blockIdx.x * blockDim.x + threadIdx.x;
  if (b >= B) return;
  float sigma = 0.1f + 0.9f * t[b];
  float r = ceilf(4.0f * sigma);
  float inv2s2 = 1.0f / (2.0f * sigma * sigma);
  float tap[TAPS];
  float s = 0.0f;
#pragma unroll
  for (int i = 0; i < TAPS; ++i) {
    float c = (float)(i - HALO);
    float v = (fabsf(c) <= r) ? __expf(-c * c * inv2s2) : 0.0f;
    tap[i] = v;
    s += v;
  }
  float inv = 1.0f / s;
#pragma unroll
  for (int i = 0; i < TAPS; ++i) w[b * TAPS + i] = tap[i] * inv;
}

__global__ void __launch_bounds__(256)
gauss_blend_kernel(const float* __restrict__ x, const float* __restrict__ t,
                   const float* __restrict__ wt, float* __restrict__ out,
                   int C, int Himg, int Wimg) {
  __shared__ __align__(16) float s_in[IN_ROWS * IN_PITCH];  // 24.3 KB
  __shared__ __align__(16) float s_h [IN_ROWS * H_PITCH];   // 23.0 KB
  __shared__ float s_w[16];

  const int tid = threadIdx.x;
  const int img = blockIdx.z;            // b*C + c
  const int b   = img / C;
  const int x0  = blockIdx.x * TILE;
  const int y0  = blockIdx.y * TILE;
  const float tb = t[b];                 // uniform -> scalar load
  const float* __restrict__ src = x   + (size_t)img * Himg * Wimg;
  float*       __restrict__ dst = out + (size_t)img * Himg * Wimg;

  if (tid < 16) s_w[tid] = (tid < TAPS) ? wt[b * TAPS + tid] : 0.0f;

  // ---- stage input tile (zero-padded halo; zero-pad == reference conv) ----
  const bool xin = (x0 >= HALO) && (x0 + TILE + HALO <= Wimg);
  if (xin) {
    // fast path: each 72-float row = 18 aligned float4 slots
    const float* rowbase = src + (x0 - HALO);
#pragma unroll
    for (int k = 0; k < 6; ++k) {
      int idx = tid + 256 * k;               // 0..1535, use 0..1439
      if (idx < IN_ROWS * 18) {
        int iy = idx / 18;
        int s4 = idx - iy * 18;
        int gy = y0 - HALO + iy;
        v4f v = {};
        if (gy >= 0 && gy < Himg)
          v = *(const v4f*)&rowbase[(size_t)gy * Wimg + 4 * s4];
        *(v4f*)&s_in[iy * IN_PITCH + 4 * s4] = v;
      }
    }
  } else {
    // x-edge blocks: per-element with zero padding
    for (int idx = tid; idx < IN_ROWS * IN_COLS; idx += 256) {
      int iy = idx / IN_COLS;
      int ix = idx - iy * IN_COLS;
      int gy = y0 - HALO + iy;
      int gx = x0 - HALO + ix;
      float v = 0.0f;
      if (gy >= 0 && gy < Himg && gx >= 0 && gx < Wimg)
        v = src[(size_t)gy * Wimg + gx];
      s_in[iy * IN_PITCH + ix] = v;
    }
  }
  __syncthreads();

  const int wave = tid >> 5;
  const int lane = tid & 31;
  const int half = lane >> 4;   // CDNA5 A/B operands split lanes 0-15 / 16-31
  const int ln   = lane & 15;

  // Banded weight operand: element (kp, n) [B role] == (m, kp) [A role]
  //   = w[kp - ln] if 0<=kp-ln<9 else 0.   A(16x4) and B(4x16) lane layouts
  // are transposes of each other, so ONE register image serves both passes.
  v2f wmat[6];
#pragma unroll
  for (int q = 0; q < 6; ++q) {
#pragma unroll
    for (int j = 0; j < 2; ++j) {
      int kp = 4 * q + 2 * half + j;
      int d  = kp - ln;
      wmat[q][j] = (d >= 0 && d < TAPS) ? s_w[d] : 0.0f;
    }
  }

  // ---- horizontal pass: Hh(16x16) = In(16x24) x Wband(24x16), 6 WMMAs ----
  for (int tIdx = wave; tIdx < 20; tIdx += 8) {      // 5 row-chunks x 4 col-chunks
    int i = tIdx >> 2;
    int c = tIdx & 3;
    const float* ain = &s_in[(16 * i + ln) * IN_PITCH + 16 * c];
    v8f acc = {};
#pragma unroll
    for (int q = 0; q < 6; ++q) {
      v2f a = *(const v2f*)&ain[4 * q + 2 * half];   // ds_load_b64 (2addr-fused)
      acc = wmma4(a, wmat[q], acc);
    }
    float* hout = &s_h[(16 * i) * H_PITCH + 16 * c + ln];
#pragma unroll
    for (int vg = 0; vg < 8; ++vg)                   // D row M = vg + 8*half
      hout[(vg + 8 * half) * H_PITCH] = acc[vg];
  }
  __syncthreads();

  // ---- vertical pass: Out(16x16) = Wband^T(16x24) x Hh(24x16), then blend
  //      IN PLACE into s_in (each interior cell is read as 'orig' by exactly
  //      the lane that overwrites it -> no cross-lane hazard) ----
  for (int tIdx = wave; tIdx < 16; tIdx += 8) {      // 4x4 output sub-tiles
    int r = tIdx >> 2;
    int c = tIdx & 3;
    const float* bin = &s_h[(16 * r) * H_PITCH + 16 * c + ln];
    v8f acc = {};
#pragma unroll
    for (int q = 0; q < 6; ++q) {
      v2f bb;
      bb[0] = bin[(4 * q + 2 * half + 0) * H_PITCH];
      bb[1] = bin[(4 * q + 2 * half + 1) * H_PITCH];
      acc = wmma4(wmat[q], bb, acc);
    }
#pragma unroll
    for (int vg = 0; vg < 8; ++vg) {
      int orow = 16 * r + vg + 8 * half;
      int ocol = 16 * c + ln;
      float* cell = &s_in[(orow + HALO) * IN_PITCH + (ocol + HALO)];
      float orig = *cell;
      *cell = fmaf(tb, acc[vg] - orig, orig);        // t*blur + (1-t)*x
    }
  }
  __syncthreads();

  // ---- vectorized writeout: s_in interior (rows 4..67, cols 4..67) ----
  const bool xout = (x0 + TILE <= Wimg);
#pragma unroll
  for (int k = 0; k < 4; ++k) {
    int idx = tid + 256 * k;                 // 0..1023 = 64 rows x 16 float4
    int oy = idx >> 4;
    int s4 = idx & 15;
    int gy = y0 + oy;
    if (gy < Himg) {
      v4f v = *(const v4f*)&s_in[(oy + HALO) * IN_PITCH + HALO + 4 * s4];
      int gx = x0 + 4 * s4;
      if (xout) {
        *(v4f*)&dst[(size_t)gy * Wimg + gx] = v;     // global_store_b128
      } else {
#pragma unroll
        for (int e = 0; e < 4; ++e)
          if (gx + e < Wimg) dst[(size_t)gy * Wimg + gx + e] = v[e];
      }
    }
  }
}

extern "C" void kernel_launch(void* const* d_in, const int* in_sizes, int n_in,
                              void* d_out, int out_size, void* d_ws, size_t ws_size,
                              hipStream_t stream) {
  const float* x1 = (const float*)d_in[0];   // (B, C, 512, 512) f32
  const float* t  = (const float*)d_in[1];   // (B,) f32
  float* out = (float*)d_out;

  const int B = in_sizes[1];
  const int Himg = 512, Wimg = 512;
  const int C = in_sizes[0] / (B * Himg * Wimg);

  float* wt = (float*)d_ws;                  // B*9 floats of taps

  gauss_taps_kernel<<<(B + 63) / 64, 64, 0, stream>>>(t, wt, B);

  dim3 grid((Wimg + TILE - 1) / TILE, (Himg + TILE - 1) / TILE, B * C);
  gauss_blend_kernel<<<grid, 256, 0, stream>>>(x1, t, wt, out, C, Himg, Wimg);
}